// PhyloDisentangler_74964359185187
// MI455X (gfx1250) — compile-verified
//
#include <hip/hip_runtime.h>
#include <hip/hip_bf16.h>

#define B_       64
#define C_IN_    256
#define CH_      128
#define NPH_     64
#define HW_      256        // R*R
#define ED_      128
#define NE_      1024
#define C_OUT_   256
#define FLAT_PH_ 16384
#define FLAT_Z_  4096
#define NROWS_   2048       // B*CB*NL

typedef __attribute__((ext_vector_type(2))) float v2f;
typedef __attribute__((ext_vector_type(8))) float v8f;

__device__ __forceinline__ float silu_f(float x) {
  // fast: x * native_rcp(1+exp(-x)) instead of IEEE divide expansion
  return __fdividef(x, 1.0f + __expf(-x));
}

__device__ __forceinline__ v8f wmma4(v2f a, v2f b, v8f c) {
  // D = A(16x4 f32) * B(4x16 f32) + C(16x16 f32)
  return __builtin_amdgcn_wmma_f32_16x16x4_f32(false, a, false, b, (short)0, c, false, false);
}

// -------- 1) silu(x) -> conv_in GEMM: M=16384 pixels, K=256, N=128 --------
__global__ void k_conv_in(const float* __restrict__ x, const float* __restrict__ w,
                          const float* __restrict__ bias, float* __restrict__ h) {
  int wave = (blockIdx.x * blockDim.x + threadIdx.x) >> 5;
  int lane = threadIdx.x & 31;
  int g = lane >> 4, r = lane & 15;
  int nt = wave & 7;        // 128/16 = 8 N tiles
  int mt = wave >> 3;       // 16384/16 = 1024 M tiles
  int mA = (mt << 4) + r;
  int bA = mA >> 8, hwA = mA & 255;
  const float* xb = x + bA * (C_IN_ * HW_) + hwA;   // x[b][c][hw], c-strided
  int n = (nt << 4) + r;
  const float* wr = w + n * C_IN_;                  // conv_in_w[n][c]
  v8f acc0 = {}, acc1 = {};
  for (int k0 = 0; k0 < C_IN_; k0 += 8) {
    int ka = k0 + 2 * g;
    v2f a, b;
    a.x = silu_f(xb[ka * HW_]);
    a.y = silu_f(xb[(ka + 1) * HW_]);
    b.x = wr[ka];
    b.y = wr[ka + 1];
    acc0 = wmma4(a, b, acc0);
    int kb = ka + 4;
    v2f a2, b2;
    a2.x = silu_f(xb[kb * HW_]);
    a2.y = silu_f(xb[(kb + 1) * HW_]);
    b2.x = wr[kb];
    b2.y = wr[kb + 1];
    acc1 = wmma4(a2, b2, acc1);
  }
  v8f acc = acc0 + acc1;
  float bo = bias[n];
#pragma unroll
  for (int v = 0; v < 8; ++v) {
    int m = (mt << 4) + v + 8 * g;
    int bb = m >> 8, hw = m & 255;
    h[bb * (CH_ * HW_) + n * HW_ + hw] = acc[v] + bo;   // h (b,128,256)
  }
}

// -------- 2) LayerNorm over phylo half (first 16384 floats of each h[b]) ----
__global__ void k_ln_stats(const float* __restrict__ h, float* __restrict__ stats) {
  __shared__ float s1[256], s2[256];
  int b = blockIdx.x;
  const float* p = h + b * (CH_ * HW_);
  float sum = 0.f, sq = 0.f;
  for (int i = threadIdx.x; i < FLAT_PH_; i += 256) { float v = p[i]; sum += v; sq += v * v; }
  s1[threadIdx.x] = sum; s2[threadIdx.x] = sq;
  __syncthreads();
  for (int s = 128; s > 0; s >>= 1) {
    if (threadIdx.x < s) { s1[threadIdx.x] += s1[threadIdx.x + s]; s2[threadIdx.x] += s2[threadIdx.x + s]; }
    __syncthreads();
  }
  if (threadIdx.x == 0) {
    float mu = s1[0] * (1.0f / FLAT_PH_);
    float var = s2[0] * (1.0f / FLAT_PH_) - mu * mu;
    stats[b * 2]     = mu;
    stats[b * 2 + 1] = rsqrtf(var + 1e-5f);
  }
}

__global__ void k_ln_apply(const float* __restrict__ h, const float* __restrict__ stats,
                           const float* __restrict__ lnw, const float* __restrict__ lnb,
                           float* __restrict__ hn) {
  int t = blockIdx.x * blockDim.x + threadIdx.x;   // B*FLAT_PH threads
  int b = t >> 14, i = t & 16383;
  float mu = stats[b * 2], rs = stats[b * 2 + 1];
  hn[t] = (h[b * (CH_ * HW_) + i] - mu) * rs * lnw[i] + lnb[i];
}

// -------- 3) big GEMM  out[m,n] = sum_k A[m,k]*W[n,k], K-split partials -----
// W is streamed once (non-temporal + prefetch); A panel stays hot in L2.
__global__ void k_gemm_nt_part(const float* __restrict__ A, const float* __restrict__ W,
                               float* __restrict__ part, int M, int N, int K,
                               int kchunk, int nsplit) {
  int wave = (blockIdx.x * blockDim.x + threadIdx.x) >> 5;
  int lane = threadIdx.x & 31;
  int g = lane >> 4, r = lane & 15;
  int ntiles = N >> 4;
  int s  = wave % nsplit;
  int t2 = wave / nsplit;
  int nt = t2 % ntiles, mt = t2 / ntiles;
  const float* Ar = A + (long)((mt << 4) + r) * K;
  const float* Wr = W + (long)((nt << 4) + r) * K;
  int kend = (s + 1) * kchunk;
  v8f acc0 = {}, acc1 = {};
  for (int k0 = s * kchunk; k0 < kend; k0 += 8) {
    if ((k0 & 31) == 0) __builtin_prefetch(Wr + k0 + 256, 0, 0);  // 1KB ahead, uniform
    int ka = k0 + 2 * g;
    v2f a, b;
    a.x = Ar[ka]; a.y = Ar[ka + 1];
    b.x = __builtin_nontemporal_load(Wr + ka);
    b.y = __builtin_nontemporal_load(Wr + ka + 1);
    acc0 = wmma4(a, b, acc0);
    int kb = ka + 4;
    v2f a2, b2;
    a2.x = Ar[kb]; a2.y = Ar[kb + 1];
    b2.x = __builtin_nontemporal_load(Wr + kb);
    b2.y = __builtin_nontemporal_load(Wr + kb + 1);
    acc1 = wmma4(a2, b2, acc1);
  }
  v8f acc = acc0 + acc1;
  float* pout = part + (long)s * M * N + ((nt << 4) + r);
#pragma unroll
  for (int v = 0; v < 8; ++v) {
    int m = (mt << 4) + v + 8 * g;
    pout[(long)m * N] = acc[v];
  }
}

__global__ void k_part_reduce(const float* __restrict__ part, const float* __restrict__ bias,
                              float* __restrict__ out, int M, int N, int nsplit) {
  int t = blockIdx.x * blockDim.x + threadIdx.x;
  if (t >= M * N) return;
  int n = t % N;
  float a = bias[n];
  for (int s = 0; s < nsplit; ++s) a += part[(long)s * M * N + t];
  out[t] = a;
}

// -------- 4) VQ: dot GEMM 2048x1024 over K=128 ------------------------------
__global__ void k_vq_dots(const float* __restrict__ z, const float* __restrict__ cb,
                          float* __restrict__ dots) {
  int wave = (blockIdx.x * blockDim.x + threadIdx.x) >> 5;
  int lane = threadIdx.x & 31;
  int g = lane >> 4, r = lane & 15;
  int nt = wave & 63;       // 1024/16 = 64 N tiles
  int mt = wave >> 6;       // 2048/16 = 128 M tiles
  int row = (mt << 4) + r;
  const float* zr = z + (row >> 5) * FLAT_Z_ + (row & 31);  // zf[row][e] = z[b][e*32+q]
  const float* cr = cb + ((nt << 4) + r) * ED_;
  v8f acc0 = {}, acc1 = {};
  for (int k0 = 0; k0 < ED_; k0 += 8) {
    int ka = k0 + 2 * g;
    v2f a, b;
    a.x = zr[ka * 32]; a.y = zr[(ka + 1) * 32];
    b.x = cr[ka];      b.y = cr[ka + 1];
    acc0 = wmma4(a, b, acc0);
    int kb = ka + 4;
    v2f a2, b2;
    a2.x = zr[kb * 32]; a2.y = zr[(kb + 1) * 32];
    b2.x = cr[kb];      b2.y = cr[kb + 1];
    acc1 = wmma4(a2, b2, acc1);
  }
  v8f acc = acc0 + acc1;
  float* dout = dots + ((nt << 4) + r);
#pragma unroll
  for (int v = 0; v < 8; ++v) {
    int m = (mt << 4) + v + 8 * g;
    dout[(long)m * NE_] = acc[v];
  }
}

__global__ void k_cnorm(const float* __restrict__ cb, float* __restrict__ cnorm) {
  int j = blockIdx.x * blockDim.x + threadIdx.x;
  if (j >= NE_) return;
  float s = 0.f;
  for (int e = 0; e < ED_; ++e) { float v = cb[j * ED_ + e]; s += v * v; }
  cnorm[j] = s;
}

__global__ void k_argmin(const float* __restrict__ dots, const float* __restrict__ cnorm,
                         int* __restrict__ idx) {
  __shared__ float sv[256]; __shared__ int si[256];
  int row = blockIdx.x;
  float best = 3.4e38f; int bi = 0;
  for (int j = threadIdx.x; j < NE_; j += 256) {
    float d = cnorm[j] - 2.0f * dots[(long)row * NE_ + j];
    if (d < best) { best = d; bi = j; }
  }
  sv[threadIdx.x] = best; si[threadIdx.x] = bi;
  __syncthreads();
  for (int s = 128; s > 0; s >>= 1) {
    if (threadIdx.x < s) {
      float ov = sv[threadIdx.x + s]; int oi = si[threadIdx.x + s];
      float mv = sv[threadIdx.x];     int mi = si[threadIdx.x];
      if (ov < mv || (ov == mv && oi < mi)) { sv[threadIdx.x] = ov; si[threadIdx.x] = oi; }
    }
    __syncthreads();
  }
  if (threadIdx.x == 0) idx[row] = si[0];
}

__global__ void k_zero(float* p) { if (threadIdx.x == 0 && blockIdx.x == 0) *p = 0.f; }

__global__ void k_gather_loss(const float* __restrict__ cb, const int* __restrict__ idx,
                              const float* __restrict__ z, float* __restrict__ zq,
                              float* __restrict__ loss_acc) {
  __shared__ float sd[256];
  int t = blockIdx.x * blockDim.x + threadIdx.x;   // 2048*128 threads
  int row = t >> 7, e = t & 127;
  int b = row >> 5, q = row & 31;
  long zi = (long)b * FLAT_Z_ + e * 32 + q;
  float zqv = cb[idx[row] * ED_ + e];
  float d = zqv - z[zi];
  zq[zi] = zqv;
  sd[threadIdx.x] = d * d;
  __syncthreads();
  for (int s = 128; s > 0; s >>= 1) {
    if (threadIdx.x < s) sd[threadIdx.x] += sd[threadIdx.x + s];
    __syncthreads();
  }
  if (threadIdx.x == 0) atomicAdd(loss_acc, sd[0]);
}

__global__ void k_loss_write(const float* __restrict__ acc, float* __restrict__ out) {
  if (threadIdx.x == 0 && blockIdx.x == 0)
    out[0] = acc[0] * 1.25f / 262144.0f;   // (1+BETA)*mean over B*ED*CB*NL
}

// -------- 5) concat + silu + conv_out GEMM: M=16384 px, K=128, N=256 --------
__global__ void k_conv_out(const float* __restrict__ hout, const float* __restrict__ h,
                           const float* __restrict__ w, const float* __restrict__ bias,
                           float* __restrict__ out) {
  int wave = (blockIdx.x * blockDim.x + threadIdx.x) >> 5;
  int lane = threadIdx.x & 31;
  int g = lane >> 4, r = lane & 15;
  int nt = wave & 15;       // 256/16 = 16 N tiles
  int mt = wave >> 4;       // 1024 M tiles
  int m = (mt << 4) + r;
  int bb = m >> 8, hw = m & 255;
  const float* hob = hout + bb * FLAT_PH_ + hw;     // channels 0..63 (c*256+hw)
  const float* hb  = h + bb * (CH_ * HW_) + hw;     // channels 64..127 live here
  int n = (nt << 4) + r;
  const float* wr = w + n * CH_;
  v8f acc0 = {}, acc1 = {};
  for (int k0 = 0; k0 < CH_; k0 += 8) {
    int ka = k0 + 2 * g;                            // even -> ka,ka+1 same region
    const float* sa = (ka < NPH_) ? hob : hb;
    v2f a, b;
    a.x = silu_f(sa[ka * HW_]);
    a.y = silu_f(sa[(ka + 1) * HW_]);
    b.x = wr[ka]; b.y = wr[ka + 1];
    acc0 = wmma4(a, b, acc0);
    int kb = ka + 4;
    const float* sb = (kb < NPH_) ? hob : hb;
    v2f a2, b2;
    a2.x = silu_f(sb[kb * HW_]);
    a2.y = silu_f(sb[(kb + 1) * HW_]);
    b2.x = wr[kb]; b2.y = wr[kb + 1];
    acc1 = wmma4(a2, b2, acc1);
  }
  v8f acc = acc0 + acc1;
  float bo = bias[n];
#pragma unroll
  for (int v = 0; v < 8; ++v) {
    int mm = (mt << 4) + v + 8 * g;
    int b2i = mm >> 8, hw2 = mm & 255;
    out[b2i * (C_OUT_ * HW_) + n * HW_ + hw2] = acc[v] + bo;
  }
}

extern "C" void kernel_launch(void* const* d_in, const int* in_sizes, int n_in,
                              void* d_out, int out_size, void* d_ws, size_t ws_size,
                              hipStream_t stream) {
  const float* x        = (const float*)d_in[0];
  const float* w_in     = (const float*)d_in[1];
  const float* b_in     = (const float*)d_in[2];
  const float* ln_w     = (const float*)d_in[3];
  const float* ln_b     = (const float*)d_in[4];
  const float* w1       = (const float*)d_in[5];   // mlp_in_w (4096,16384)
  const float* b1       = (const float*)d_in[6];
  const float* codebook = (const float*)d_in[7];   // (1024,128)
  const float* w2       = (const float*)d_in[8];   // mlp_out_w (16384,4096)
  const float* b2       = (const float*)d_in[9];
  const float* w_out    = (const float*)d_in[10];
  const float* b_out    = (const float*)d_in[11];
  float* out = (float*)d_out;
  float* ws  = (float*)d_ws;

  // workspace layout (float offsets)
  const long OFF_H     = 0;          // 2,097,152  h (b,128,256)
  const long OFF_HN    = 2097152;    // 1,048,576  normalized phylo
  const long OFF_STATS = 3145728;    // 128        (mu, rsig) per batch
  const long OFF_Z     = 3145856;    // 262,144    z (b,4096)
  const long OFF_ZQ    = 3408000;    // 262,144    zq (b,4096)
  const long OFF_CN    = 3670144;    // 1,024      codebook norms
  const long OFF_IDX   = 3671168;    // 2,048      argmin indices (int)
  const long OFF_LOSS  = 3673216;    // 1          loss accumulator
  const long OFF_HOUT  = 3673344;    // 1,048,576  mlp_out result
  const long OFF_PART  = 4721920;    // 4,194,304  reused: z-partials / dots / hout-partials

  // 1) silu + conv_in (8192 waves)
  k_conv_in<<<1024, 256, 0, stream>>>(x, w_in, b_in, ws + OFF_H);
  // 2) layernorm
  k_ln_stats<<<64, 256, 0, stream>>>(ws + OFF_H, ws + OFF_STATS);
  k_ln_apply<<<4096, 256, 0, stream>>>(ws + OFF_H, ws + OFF_STATS, ln_w, ln_b, ws + OFF_HN);
  // 3) mlp_in: M=64 N=4096 K=16384, 8-way K split (8192 waves)
  k_gemm_nt_part<<<1024, 256, 0, stream>>>(ws + OFF_HN, w1, ws + OFF_PART, 64, 4096, 16384, 2048, 8);
  k_part_reduce<<<1024, 256, 0, stream>>>(ws + OFF_PART, b1, ws + OFF_Z, 64, 4096, 8);
  // 4) vector quantize
  k_cnorm<<<4, 256, 0, stream>>>(codebook, ws + OFF_CN);
  k_vq_dots<<<1024, 256, 0, stream>>>(ws + OFF_Z, codebook, ws + OFF_PART);
  k_argmin<<<2048, 256, 0, stream>>>(ws + OFF_PART, ws + OFF_CN, (int*)(ws + OFF_IDX));
  k_zero<<<1, 1, 0, stream>>>(ws + OFF_LOSS);
  k_gather_loss<<<1024, 256, 0, stream>>>(codebook, (const int*)(ws + OFF_IDX),
                                          ws + OFF_Z, ws + OFF_ZQ, ws + OFF_LOSS);
  // 5) mlp_out: M=64 N=16384 K=4096, 4-way K split (16384 waves)
  k_gemm_nt_part<<<2048, 256, 0, stream>>>(ws + OFF_ZQ, w2, ws + OFF_PART, 64, 16384, 4096, 1024, 4);
  k_part_reduce<<<4096, 256, 0, stream>>>(ws + OFF_PART, b2, ws + OFF_HOUT, 64, 16384, 4);
  // 6) concat + silu + conv_out -> d_out
  k_conv_out<<<2048, 256, 0, stream>>>(ws + OFF_HOUT, ws + OFF_H, w_out, b_out, out);
  // 7) q_loss -> d_out[4194304]
  k_loss_write<<<1, 1, 0, stream>>>(ws + OFF_LOSS, out + 4194304);
}